// GCNLayer_25890062861001
// MI455X (gfx1250) — compile-verified
//
#include <hip/hip_runtime.h>

// GCNConv forward: out = D^{-1/2}(A+I)D^{-1/2} (x W) + b
// MI455X plan: GEMM via V_WMMA_F32_16X16X4_F32 (exact fp32), scatter via
// L2-resident f32 atomics (h and out both fit in 192MB L2).

typedef __attribute__((ext_vector_type(2))) float v2f;
typedef __attribute__((ext_vector_type(8))) float v8f;

#define DD 128  // D_IN == D_OUT == 128

// ---- kernel 1: deg = 1 (self loop) -----------------------------------------
__global__ void k_init_deg(float* __restrict__ deg, int n) {
  int i = blockIdx.x * blockDim.x + threadIdx.x;
  if (i < n) deg[i] = 1.0f;
}

// ---- kernel 2: deg[col] += 1 per edge --------------------------------------
__global__ void k_count_deg(const int* __restrict__ cols, float* __restrict__ deg, int e) {
  int i = blockIdx.x * blockDim.x + threadIdx.x;
  if (i < e) atomicAdd(&deg[cols[i]], 1.0f);
}

// ---- kernel 3: h = x @ W via WMMA f32 16x16x4 ------------------------------
// Block = 256 threads = 8 waves. W (128x128 f32 = 64KB) staged in LDS once.
// Wave w computes the 16-column tile n0 = 16*w for each 16-row tile (grid-stride).
__global__ void k_gemm_wmma(const float* __restrict__ x, const float* __restrict__ W,
                            float* __restrict__ h, int n, int n_tiles) {
  __shared__ float Wlds[DD * DD];
  int t = threadIdx.x;
  for (int i = t; i < DD * DD; i += 256) Wlds[i] = W[i];
  __syncthreads();

  int lane  = t & 31;
  int wave  = t >> 5;        // 0..7
  int n0    = wave * 16;
  int l15   = lane & 15;
  int khalf = lane >> 4;     // 0 or 1 (lane-half selects K pair / M half)

  for (int tile = blockIdx.x; tile < n_tiles; tile += gridDim.x) {
    int m0 = tile * 16;
    int m  = m0 + l15; if (m >= n) m = n - 1;   // clamp (EXEC must stay all-1s for WMMA)
    const float* xa = x + (size_t)m * DD + 2 * khalf;

    v8f c = {};
#pragma unroll
    for (int kb = 0; kb < DD / 4; ++kb) {
      int k0 = kb * 4;
      // A 16x4 f32 layout: lanes 0-15 hold K={k0,k0+1}, lanes 16-31 hold K={k0+2,k0+3}
      v2f a; a.x = xa[k0]; a.y = xa[k0 + 1];
      // B 4x16 f32 layout: VGPR j holds row K=k0+j (lanes 0-15) / K=k0+j+2 (lanes 16-31)
      v2f bf;
      bf.x = Wlds[(k0 + 2 * khalf) * DD + n0 + l15];
      bf.y = Wlds[(k0 + 2 * khalf + 1) * DD + n0 + l15];
      // 8 args: (neg_a, A, neg_b, B, c_mod, C, reuse_a, reuse_b)
      c = __builtin_amdgcn_wmma_f32_16x16x4_f32(false, a, false, bf, (short)0, c, false, false);
    }

    // C/D 16x16 f32 layout: VGPR r -> row m0 + r + 8*khalf, col n0 + l15
#pragma unroll
    for (int r = 0; r < 8; ++r) {
      int row = m0 + r + 8 * khalf;
      if (row < n) h[(size_t)row * DD + n0 + l15] = c[r];
    }
  }
}

// ---- kernel 4: dis = rsqrt(deg); out = b + dis^2 * h (self-loop term) ------
__global__ void k_selfloop(const float* __restrict__ h, const float* __restrict__ deg,
                           const float* __restrict__ bias, float* __restrict__ dis,
                           float* __restrict__ out, int n) {
  long gid = (long)blockIdx.x * blockDim.x + threadIdx.x;
  int i = (int)(gid >> 7);
  int d = (int)(gid & 127);
  if (i >= n) return;
  float s = rsqrtf(deg[i]);          // deg >= 1 always (self loop)
  if (d == 0) dis[i] = s;
  out[(size_t)i * DD + d] = bias[d] + s * s * h[(size_t)i * DD + d];
}

// ---- kernel 5: edge scatter-add: out[col] += dis[row]*dis[col] * h[row] ----
// One wave32 per edge; each lane moves a float4 (128 floats = 32 lanes * 4).
__global__ void k_scatter(const int* __restrict__ rows, const int* __restrict__ cols,
                          const float* __restrict__ dis, const float* __restrict__ h,
                          float* __restrict__ out, int e) {
  long gid = (long)blockIdx.x * blockDim.x + threadIdx.x;
  int eidx = (int)(gid >> 5);
  int lane = (int)(gid & 31);
  if (eidx >= e) return;
  int r = rows[eidx];
  int c = cols[eidx];
  float norm = dis[r] * dis[c];
  float4 m = *((const float4*)(h + (size_t)r * DD) + lane);
  float* op = out + (size_t)c * DD + lane * 4;
  atomicAdd(op + 0, norm * m.x);
  atomicAdd(op + 1, norm * m.y);
  atomicAdd(op + 2, norm * m.z);
  atomicAdd(op + 3, norm * m.w);
}

extern "C" void kernel_launch(void* const* d_in, const int* in_sizes, int n_in,
                              void* d_out, int out_size, void* d_ws, size_t ws_size,
                              hipStream_t stream) {
  const float* x    = (const float*)d_in[0];
  const int*   ei   = (const int*)  d_in[1];   // [2, E] flat: rows then cols
  const float* W    = (const float*)d_in[2];
  const float* bias = (const float*)d_in[3];
  float* out = (float*)d_out;

  int n = in_sizes[0] / DD;
  int e = in_sizes[1] / 2;
  const int* rows = ei;
  const int* cols = ei + e;

  // workspace: h [n*128], deg [n], dis [n]
  float* h   = (float*)d_ws;
  float* deg = h + (size_t)n * DD;
  float* dis = deg + n;

  k_init_deg<<<(n + 255) / 256, 256, 0, stream>>>(deg, n);
  k_count_deg<<<(e + 255) / 256, 256, 0, stream>>>(cols, deg, e);

  int n_tiles = (n + 15) / 16;
  int gblocks = n_tiles < 2048 ? n_tiles : 2048;
  k_gemm_wmma<<<gblocks, 256, 0, stream>>>(x, W, h, n, n_tiles);

  long nt = (long)n * DD;
  k_selfloop<<<(int)((nt + 255) / 256), 256, 0, stream>>>(h, deg, bias, dis, out, n);

  long st = (long)e * 32;
  k_scatter<<<(int)((st + 255) / 256), 256, 0, stream>>>(rows, cols, dis, h, out, e);
}